// GNNlayer_14499809591715
// MI455X (gfx1250) — compile-verified
//
#include <hip/hip_runtime.h>
#include <hip/hip_bf16.h>
#include <math.h>

// ---------------------------------------------------------------------------
// GNN layer: S = lap(A); P = I + t0*S + t1*S^2; S2 = lap(P); top-4/row mask;
// y = S2_masked @ x; out = y @ W^T + b.
// Dominant cost: fp32 4096^3 GEMM -> V_WMMA_F32_16X16X4_F32 (fp32 matrix path;
// bf16 would perturb the top-k ranking). A (64MB) / S (64MB) / P (64MB) are
// L2-resident on MI455X (192MB L2) -> GEMM is compute-bound. Panel staging
// uses GLOBAL_LOAD_ASYNC_TO_LDS_B128 double-buffering when available.
// ---------------------------------------------------------------------------

typedef float v2f __attribute__((ext_vector_type(2)));
typedef float v8f __attribute__((ext_vector_type(8)));
typedef int   v4i __attribute__((ext_vector_type(4)));

static constexpr int N    = 4096;
static constexpr int DIN  = 64;
static constexpr int DOUT = 64;
static constexpr int TOPK = 4;   // reference k == 4

// ----------------------- CDNA5 async-to-LDS helpers ------------------------
#if defined(__has_builtin)
#if __has_builtin(__builtin_amdgcn_global_load_async_to_lds_b128)
#define HAVE_ASYNC_LDS 1
#endif
#endif
#ifndef HAVE_ASYNC_LDS
#define HAVE_ASYNC_LDS 0
#endif

__device__ __forceinline__ void async_copy16(const float* g, float* l) {
#if HAVE_ASYNC_LDS
  // param0: int4 __device__* (global, non-const); param1: LDS-side pointer
  __builtin_amdgcn_global_load_async_to_lds_b128(
      (__attribute__((address_space(1))) v4i*)g,
      (__attribute__((address_space(3))) v4i*)l,
      /*offset=*/0, /*cpol=*/0);
#else
  *(float4*)l = *(const float4*)g;
#endif
}

__device__ __forceinline__ void wait_async() {
#if HAVE_ASYNC_LDS
#if __has_builtin(__builtin_amdgcn_s_wait_asynccnt)
  __builtin_amdgcn_s_wait_asynccnt(0);
#else
  asm volatile("s_wait_asynccnt 0x0" ::: "memory");
#endif
#endif
}

// ---------------- K1 / K3: row sums -> dinv = rsqrt(rowsum) ----------------
__global__ __launch_bounds__(256)
void rowsum_rsqrt_kernel(const float* __restrict__ M, float* __restrict__ dinv) {
  __shared__ float red[256];
  const int row = blockIdx.x;
  const int tid = threadIdx.x;
  const float* r = M + (size_t)row * N;
  float s = 0.0f;
  for (int j = tid; j < N; j += 256) s += r[j];
  red[tid] = s;
  __syncthreads();
  for (int off = 128; off > 0; off >>= 1) {
    if (tid < off) red[tid] += red[tid + off];
    __syncthreads();
  }
  if (tid == 0) dinv[row] = rsqrtf(red[0]);
}

// ---------------- K1.5: S[i][j] = A[i][j] * dinv[i] * dinv[j] --------------
__global__ __launch_bounds__(256)
void scale_S_kernel(const float* __restrict__ A, const float* __restrict__ dinv,
                    float* __restrict__ S) {
  const int row = blockIdx.x;
  const int tid = threadIdx.x;
  const float di = dinv[row];
  const float4* a4 = (const float4*)(A + (size_t)row * N);
  const float4* d4 = (const float4*)dinv;
  float4* s4 = (float4*)(S + (size_t)row * N);
  for (int j = tid; j < N / 4; j += 256) {
    float4 a = a4[j], d = d4[j];
    float4 o;
    o.x = a.x * di * d.x; o.y = a.y * di * d.y;
    o.z = a.z * di * d.z; o.w = a.w * di * d.w;
    s4[j] = o;
  }
}

// ======================= GEMM tiling (both variants) =======================
// Block tile 128x128, 8 waves (4 along M x 2 along N), each wave 32x64
// (2x4 tiles of 16x16), BK=16 (4 WMMA k-steps per staging round).
static constexpr int BM = 128, BN = 128, BK = 16;

// ---------- K2a: prescaled GEMM, async double-buffered staging -------------
// sA m-major, row stride 20 floats: 16B-aligned rows for B128 async writes,
// and ds_load_b64 frag reads hit banks 20*l mod 64 (all distinct) -> no
// conflicts; A frag K-pair is contiguous -> single b64 per frag.
static constexpr int LDAP = BK + 4;   // 20

__global__ __launch_bounds__(256)
void gemm_poly_prescaled_kernel(const float* __restrict__ S,
                                const float* __restrict__ theta,
                                float* __restrict__ P) {
  __shared__ float sA[2][BM * LDAP];  // sA[b][m*LDAP + k] = S[bm0+m][k0+k]
  __shared__ float sB[2][BK * BN];    // sB[b][k*BN + n]   = S[k0+k][bn0+n]

  const int tid  = threadIdx.x;
  const int lane = tid & 31;
  const int wave = tid >> 5;
  const int half = lane >> 4;
  const int l16  = lane & 15;
  const int wm0  = (wave & 3) * 32;
  const int wn0  = (wave >> 2) * 64;
  const int bm0  = blockIdx.x * BM;
  const int bn0  = blockIdx.y * BN;

  v8f acc[2][4];
  #pragma unroll
  for (int mt = 0; mt < 2; ++mt)
    #pragma unroll
    for (int nt = 0; nt < 4; ++nt)
      #pragma unroll
      for (int v = 0; v < 8; ++v) acc[mt][nt][v] = 0.0f;

  // stage one A panel (128x16) + one B panel (16x128) = 512 16B chunks each
  auto stage = [&](float* dA, float* dB, int k0) {
    #pragma unroll
    for (int c = tid; c < 512; c += 256) {
      const int m = c >> 2, k4 = (c & 3) << 2;       // A: 4 chunks per row
      async_copy16(S + (size_t)(bm0 + m) * N + k0 + k4, dA + m * LDAP + k4);
      const int kb = c >> 5, n4 = (c & 31) << 2;     // B: 32 chunks per k-row
      async_copy16(S + (size_t)(k0 + kb) * N + bn0 + n4, dB + kb * BN + n4);
    }
  };

  stage(sA[0], sB[0], 0);
  wait_async();
  __syncthreads();

  for (int k0 = 0; k0 < N; k0 += BK) {
    const int cur = (k0 >> 4) & 1;
    const int nxt = cur ^ 1;
    if (k0 + BK < N) stage(sA[nxt], sB[nxt], k0 + BK);  // overlap DMA w/ WMMA

    const float* sa = sA[cur];
    const float* sb = sB[cur];
    #pragma unroll
    for (int kk = 0; kk < BK; kk += 4) {
      // A 16x4 frag: VGPR0 = K=kk(+2*half), VGPR1 = K=kk+1(+2*half), M=l16
      v2f afrag[2];
      #pragma unroll
      for (int mt = 0; mt < 2; ++mt)
        afrag[mt] = *(const v2f*)&sa[(wm0 + mt * 16 + l16) * LDAP + kk + 2 * half];
      // B 4x16 frag: VGPR v holds K=kk+v (lanes0-15) / K=kk+v+2 (lanes16-31)
      v2f bfrag[4];
      #pragma unroll
      for (int nt = 0; nt < 4; ++nt) {
        const int n = wn0 + nt * 16 + l16;
        bfrag[nt].x = sb[(kk + 2 * half + 0) * BN + n];
        bfrag[nt].y = sb[(kk + 2 * half + 1) * BN + n];
      }
      #pragma unroll
      for (int mt = 0; mt < 2; ++mt)
        #pragma unroll
        for (int nt = 0; nt < 4; ++nt)
          acc[mt][nt] = __builtin_amdgcn_wmma_f32_16x16x4_f32(
              false, afrag[mt], false, bfrag[nt],
              (short)0, acc[mt][nt], false, false);
    }
    wait_async();
    __syncthreads();
  }

  // epilogue: P = I + t0*S + t1*(S@S)
  const float t0 = 1.0f / (1.0f + expf(-theta[0]));
  const float t1 = 1.0f / (1.0f + expf(-theta[1]));
  #pragma unroll
  for (int nt = 0; nt < 4; ++nt) {
    const int gj = bn0 + wn0 + nt * 16 + l16;
    #pragma unroll
    for (int mt = 0; mt < 2; ++mt) {
      #pragma unroll
      for (int v = 0; v < 8; ++v) {
        const int gi = bm0 + wm0 + mt * 16 + v + 8 * half;  // C/D: M=v+8*half
        float val = t1 * acc[mt][nt][v] + t0 * S[(size_t)gi * N + gj];
        if (gi == gj) val += 1.0f;
        P[(size_t)gi * N + gj] = val;
      }
    }
  }
}

// ---------- K2b: fallback fused GEMM (small workspace), scaling in staging --
static constexpr int LDA_S = BM + 1;
static constexpr int LDB_S = BN + 1;

__global__ __launch_bounds__(256)
void gemm_poly_fused_kernel(const float* __restrict__ A,
                            const float* __restrict__ dinv1,
                            const float* __restrict__ theta,
                            float* __restrict__ P) {
  __shared__ float sA[BK * LDA_S];  // sA[k][m] = A[bm0+m][k0+k] * dinv1[k0+k]
  __shared__ float sB[BK * LDB_S];  // sB[k][n] = A[k0+k][bn0+n] * dinv1[k0+k]

  const int tid  = threadIdx.x;
  const int lane = tid & 31;
  const int wave = tid >> 5;
  const int half = lane >> 4;
  const int l16  = lane & 15;
  const int wm0  = (wave & 3) * 32;
  const int wn0  = (wave >> 2) * 64;
  const int bm0  = blockIdx.x * BM;
  const int bn0  = blockIdx.y * BN;

  v8f acc[2][4];
  #pragma unroll
  for (int mt = 0; mt < 2; ++mt)
    #pragma unroll
    for (int nt = 0; nt < 4; ++nt)
      #pragma unroll
      for (int v = 0; v < 8; ++v) acc[mt][nt][v] = 0.0f;

  const int am  = tid >> 1;
  const int ak0 = (tid & 1) * 8;
  const int bk  = tid >> 4;
  const int bn  = (tid & 15) * 8;

  for (int k0 = 0; k0 < N; k0 += BK) {
    const float* gA = A + (size_t)(bm0 + am) * N + (k0 + ak0);
    #pragma unroll
    for (int i = 0; i < 8; ++i)
      sA[(ak0 + i) * LDA_S + am] = gA[i] * dinv1[k0 + ak0 + i];
    const float dk = dinv1[k0 + bk];
    const float* gB = A + (size_t)(k0 + bk) * N + (bn0 + bn);
    #pragma unroll
    for (int i = 0; i < 8; ++i)
      sB[bk * LDB_S + bn + i] = gB[i] * dk;
    __syncthreads();

    #pragma unroll
    for (int kk = 0; kk < BK; kk += 4) {
      v2f afrag[2];
      #pragma unroll
      for (int mt = 0; mt < 2; ++mt) {
        const int m = wm0 + mt * 16 + l16;
        afrag[mt].x = sA[(kk + 2 * half + 0) * LDA_S + m];
        afrag[mt].y = sA[(kk + 2 * half + 1) * LDA_S + m];
      }
      v2f bfrag[4];
      #pragma unroll
      for (int nt = 0; nt < 4; ++nt) {
        const int n = wn0 + nt * 16 + l16;
        bfrag[nt].x = sB[(kk + 2 * half + 0) * LDB_S + n];
        bfrag[nt].y = sB[(kk + 2 * half + 1) * LDB_S + n];
      }
      #pragma unroll
      for (int mt = 0; mt < 2; ++mt)
        #pragma unroll
        for (int nt = 0; nt < 4; ++nt)
          acc[mt][nt] = __builtin_amdgcn_wmma_f32_16x16x4_f32(
              false, afrag[mt], false, bfrag[nt],
              (short)0, acc[mt][nt], false, false);
    }
    __syncthreads();
  }

  const float t0 = 1.0f / (1.0f + expf(-theta[0]));
  const float t1 = 1.0f / (1.0f + expf(-theta[1]));
  #pragma unroll
  for (int nt = 0; nt < 4; ++nt) {
    const int gj = bn0 + wn0 + nt * 16 + l16;
    const float dj = dinv1[gj];
    #pragma unroll
    for (int mt = 0; mt < 2; ++mt) {
      #pragma unroll
      for (int v = 0; v < 8; ++v) {
        const int gi = bm0 + wm0 + mt * 16 + v + 8 * half;
        const float s = dinv1[gi] * dj;
        float val = t1 * s * acc[mt][nt][v] + t0 * s * A[(size_t)gi * N + gj];
        if (gi == gj) val += 1.0f;
        P[(size_t)gi * N + gj] = val;
      }
    }
  }
}

// ------------- K4: per-row top-4 (stable-argsort semantics) + gather -------
__global__ __launch_bounds__(256)
void topk_gather_kernel(const float* __restrict__ P,
                        const float* __restrict__ dinv2,
                        const float* __restrict__ x,
                        float* __restrict__ y) {
  __shared__ float svals[256 * TOPK];
  __shared__ int   sidx [256 * TOPK];
  __shared__ float fvals[TOPK];
  __shared__ int   fidx [TOPK];

  const int row = blockIdx.x;
  const int tid = threadIdx.x;
  const float* prow = P + (size_t)row * N;

  float bv[TOPK];
  int   bi[TOPK];
  #pragma unroll
  for (int m = 0; m < TOPK; ++m) { bv[m] = -__builtin_inff(); bi[m] = 0x7fffffff; }
  for (int j = tid; j < N; j += 256) {
    const float v = prow[j] * dinv2[j];
    if (v > bv[TOPK - 1]) {          // strict > : equal values keep earlier index
      bv[TOPK - 1] = v; bi[TOPK - 1] = j;
      #pragma unroll
      for (int p = TOPK - 1; p > 0; --p) {
        if (bv[p] > bv[p - 1]) {
          float tv = bv[p]; bv[p] = bv[p - 1]; bv[p - 1] = tv;
          int   ti = bi[p]; bi[p] = bi[p - 1]; bi[p - 1] = ti;
        }
      }
    }
  }
  #pragma unroll
  for (int m = 0; m < TOPK; ++m) { svals[tid * TOPK + m] = bv[m]; sidx[tid * TOPK + m] = bi[m]; }
  __syncthreads();

  float mv[TOPK]; int mi[TOPK];
  #pragma unroll
  for (int m = 0; m < TOPK; ++m) { mv[m] = -__builtin_inff(); mi[m] = 0x7fffffff; }
  if (tid < 32) {
    for (int s = tid * 32; s < tid * 32 + 32; ++s) {
      const float v = svals[s]; const int j = sidx[s];
      if (v > mv[TOPK - 1] || (v == mv[TOPK - 1] && j < mi[TOPK - 1])) {
        mv[TOPK - 1] = v; mi[TOPK - 1] = j;
        #pragma unroll
        for (int p = TOPK - 1; p > 0; --p) {
          if (mv[p] > mv[p - 1] || (mv[p] == mv[p - 1] && mi[p] < mi[p - 1])) {
            float tv = mv[p]; mv[p] = mv[p - 1]; mv[p - 1] = tv;
            int   ti = mi[p]; mi[p] = mi[p - 1]; mi[p - 1] = ti;
          }
        }
      }
    }
  }
  __syncthreads();
  if (tid < 32) {
    #pragma unroll
    for (int m = 0; m < TOPK; ++m) { svals[tid * TOPK + m] = mv[m]; sidx[tid * TOPK + m] = mi[m]; }
  }
  __syncthreads();

  if (tid == 0) {
    float gv[TOPK]; int gi[TOPK];
    #pragma unroll
    for (int m = 0; m < TOPK; ++m) { gv[m] = -__builtin_inff(); gi[m] = 0x7fffffff; }
    for (int s = 0; s < 32 * TOPK; ++s) {
      const float v = svals[s]; const int j = sidx[s];
      if (v > gv[TOPK - 1] || (v == gv[TOPK - 1] && j < gi[TOPK - 1])) {
        gv[TOPK - 1] = v; gi[TOPK - 1] = j;
        #pragma unroll
        for (int p = TOPK - 1; p > 0; --p) {
          if (gv[p] > gv[p - 1] || (gv[p] == gv[p - 1] && gi[p] < gi[p - 1])) {
            float tv = gv[p]; gv[p] = gv[p - 1]; gv[p - 1] = tv;
            int   ti = gi[p]; gi[p] = gi[p - 1]; gi[p - 1] = ti;
          }
        }
      }
    }
    const float di = dinv2[row];   // kept value = P * dinv2[row] * dinv2[j]
    #pragma unroll
    for (int m = 0; m < TOPK; ++m) { fvals[m] = gv[m] * di; fidx[m] = gi[m]; }
  }
  __syncthreads();

  if (tid < DIN) {
    float a = 0.0f;
    #pragma unroll
    for (int m = 0; m < TOPK; ++m) a += fvals[m] * x[(size_t)fidx[m] * DIN + tid];
    y[(size_t)row * DIN + tid] = a;
  }
}

// ---------------- K5: out = y @ W^T + b (W async-staged in LDS) ------------
__global__ __launch_bounds__(256)
void linear_kernel(const float* __restrict__ y, const float* __restrict__ W,
                   const float* __restrict__ b, float* __restrict__ out) {
  __shared__ float sW[DOUT * DIN];
  __shared__ float sb[DOUT];
  const int tid = threadIdx.x;
  #pragma unroll
  for (int c = tid; c < (DOUT * DIN) / 4; c += 256)
    async_copy16(W + c * 4, sW + c * 4);
  if (tid < DOUT / 4) async_copy16(b + tid * 4, sb + tid * 4);
  wait_async();
  __syncthreads();
  const int gid = blockIdx.x * 256 + tid;
  const int row = gid >> 6;       // DIN == DOUT == 64
  const int o   = gid & 63;
  const float* yr = y + (size_t)row * DIN;
  float acc = sb[o];
  #pragma unroll
  for (int d = 0; d < DIN; ++d) acc += yr[d] * sW[o * DIN + d];
  out[gid] = acc;
}

// ---------------------------------------------------------------------------
extern "C" void kernel_launch(void* const* d_in, const int* in_sizes, int n_in,
                              void* d_out, int out_size, void* d_ws, size_t ws_size,
                              hipStream_t stream) {
  (void)in_sizes; (void)n_in; (void)out_size;
  const float* x     = (const float*)d_in[0];  // [N, DIN]
  const float* A     = (const float*)d_in[1];  // [N, N]
  const float* theta = (const float*)d_in[2];  // [2]
  const float* W     = (const float*)d_in[3];  // [DOUT, DIN]
  const float* b     = (const float*)d_in[4];  // [DOUT]
  // d_in[5] = k (== 4, baked in as TOPK)

  float* ws    = (float*)d_ws;
  float* dinv1 = ws;                        // 4096
  float* dinv2 = ws + N;                    // 4096
  float* y     = ws + 2 * N;                // N*DIN
  float* P     = ws + 2 * N + N * DIN;      // N*N
  float* out   = (float*)d_out;             // [N, DOUT]

  const size_t big_need =
      ((size_t)2 * N + (size_t)N * DIN + 2 * (size_t)N * N) * sizeof(float);
  const dim3 grid(N / BM, N / BN);

  rowsum_rsqrt_kernel<<<N, 256, 0, stream>>>(A, dinv1);
  if (ws_size >= big_need) {
    // room for both S and P: prescale S, run pure async-staged S@S GEMM
    float* S = P + (size_t)N * N;
    scale_S_kernel<<<N, 256, 0, stream>>>(A, dinv1, S);
    gemm_poly_prescaled_kernel<<<grid, 256, 0, stream>>>(S, theta, P);
  } else {
    gemm_poly_fused_kernel<<<grid, 256, 0, stream>>>(A, dinv1, theta, P);
  }
  rowsum_rsqrt_kernel<<<N, 256, 0, stream>>>(P, dinv2);
  topk_gather_kernel<<<N, 256, 0, stream>>>(P, dinv2, x, y);
  linear_kernel<<<(N * DOUT) / 256, 256, 0, stream>>>(y, W, b, out);
}